// BiggerBirdBartForSequenceClassification_19224273617380
// MI455X (gfx1250) — compile-verified
//
#include <hip/hip_runtime.h>
#include <math.h>

typedef float v2f __attribute__((ext_vector_type(2)));
typedef float v8f __attribute__((ext_vector_type(8)));

#define B_    2
#define H_    12
#define T_    2048
#define D_    64
#define BH_   (B_ * H_)
#define FRAG_ 32
#define KSEL_ 6
#define G_    3
#define KC_   24          // min(32, max(6, ceil(4*6)))
#define STEPS_ 3          // min(KSEL, 1 + DIV_STEPS)
#define P_    16
#define TOPU_ 8
#define KQ_   3           // max(1, round(16*0.2))
#define TAU_   8.0f
#define ALPHA_ 0.2f
#define GAMMA_ 0.2f
#define SCALE_ 0.125f     // 1/sqrt(64)
#define NEG_   (-1.0e9f)
#define W_MEAN_ 1.0f
#define W_MAX_  0.6f
#define W_TOPK_ 0.4f
#define W_STD_  0.2f
#define TILE_ 16
#define UK_   48          // union keys per 16-row tile (15 shift + 32 window <= 48)

// ---------------------------------------------------------------------------
// Kernel 1: per-head global-key selection (_choose_globals). One block / head.
// ---------------------------------------------------------------------------
__global__ void bbird_choose_globals(const float* __restrict__ q,
                                     const float* __restrict__ k,
                                     int* __restrict__ g_sel) {
  __shared__ float Qp[P_][D_];      // normalized probe queries
  __shared__ float u[T_];           // per-key utility (8 KB)
  __shared__ float Ssub[TOPU_][P_];
  __shared__ int   topIdx[TOPU_];

  const int h = blockIdx.x;
  const int tid = threadIdx.x;
  const int idxp[P_] = {0,136,273,409,546,682,819,955,
                        1092,1228,1365,1501,1638,1774,1911,2047};

  // --- probe queries: normalize(mean over batch of q) at idxp positions ---
  if (tid < P_) {
    const int tp = idxp[tid];
    const float* q0 = q + ((size_t)(h * B_ + 0) * T_ + tp) * D_;
    const float* q1 = q + ((size_t)(h * B_ + 1) * T_ + tp) * D_;
    float tmp[D_]; float ss = 0.f;
    for (int d = 0; d < D_; ++d) { float m = 0.5f * (q0[d] + q1[d]); tmp[d] = m; ss += m * m; }
    const float rn = 1.0f / fmaxf(sqrtf(ss), 1e-6f);
    for (int d = 0; d < D_; ++d) Qp[tid][d] = tmp[d] * rn;
  }
  __syncthreads();

  // --- utility per key t ---
  for (int t = tid; t < T_; t += blockDim.x) {
    const float* k0 = k + ((size_t)(h * B_ + 0) * T_ + t) * D_;
    const float* k1 = k + ((size_t)(h * B_ + 1) * T_ + t) * D_;
    float kb[D_]; float ss = 0.f;
    for (int d = 0; d < D_; ++d) { float m = 0.5f * (k0[d] + k1[d]); kb[d] = m; ss += m * m; }
    const float rn = 1.0f / fmaxf(sqrtf(ss), 1e-6f);
    float s[P_]; float mean = 0.f, mx = -1e30f;
    for (int p = 0; p < P_; ++p) {
      float dot = 0.f;
      for (int d = 0; d < D_; ++d) dot += kb[d] * Qp[p][d];
      float sv = fmaxf(dot * rn, 0.f);
      s[p] = sv; mean += sv; mx = fmaxf(mx, sv);
    }
    mean *= (1.0f / P_);
    float cpy[P_];
    for (int p = 0; p < P_; ++p) cpy[p] = s[p];
    float tk = 0.f;
    for (int r = 0; r < KQ_; ++r) {
      int jb = 0; float vb = cpy[0];
      for (int p = 1; p < P_; ++p) if (cpy[p] > vb) { vb = cpy[p]; jb = p; }
      tk += vb; cpy[jb] = -1e30f;
    }
    tk *= (1.0f / KQ_);
    float sq = 0.f;
    for (int p = 0; p < P_; ++p) { float dl = s[p] - mean; sq += dl * dl; }
    const float sd = sqrtf(sq / (float)(P_ - 1));
    u[t] = W_MEAN_ * mean + W_MAX_ * mx + W_TOPK_ * tk + W_STD_ * sd;
  }
  __syncthreads();

  // --- top-8 keys by utility (stable: first max wins, matches lax.top_k) ---
  if (tid == 0) {
    for (int r = 0; r < TOPU_; ++r) {
      int jb = 0; float vb = u[0];
      for (int t = 1; t < T_; ++t) if (u[t] > vb) { vb = u[t]; jb = t; }
      topIdx[r] = jb; u[jb] = -1e30f;
    }
  }
  __syncthreads();

  // --- recompute S rows for the 8 selected keys ---
  if (tid < TOPU_) {
    const int t = topIdx[tid];
    const float* k0 = k + ((size_t)(h * B_ + 0) * T_ + t) * D_;
    const float* k1 = k + ((size_t)(h * B_ + 1) * T_ + t) * D_;
    float kb[D_]; float ss = 0.f;
    for (int d = 0; d < D_; ++d) { float m = 0.5f * (k0[d] + k1[d]); kb[d] = m; ss += m * m; }
    const float rn = 1.0f / fmaxf(sqrtf(ss), 1e-6f);
    for (int p = 0; p < P_; ++p) {
      float dot = 0.f;
      for (int d = 0; d < D_; ++d) dot += kb[d] * Qp[p][d];
      Ssub[tid][p] = fmaxf(dot * rn, 0.f);
    }
  }
  __syncthreads();

  // --- greedy facility-location pick of 3 ---
  if (tid == 0) {
    float m[P_]; bool blocked[TOPU_];
    for (int p = 0; p < P_; ++p) m[p] = 0.f;
    for (int j = 0; j < TOPU_; ++j) blocked[j] = false;
    for (int g = 0; g < G_; ++g) {
      int jb = 0; float vb = -2e30f;
      for (int j = 0; j < TOPU_; ++j) {
        float gain;
        if (blocked[j]) gain = NEG_;
        else {
          gain = 0.f;
          for (int p = 0; p < P_; ++p) gain += fmaxf(Ssub[j][p] - m[p], 0.f);
        }
        if (gain > vb) { vb = gain; jb = j; }
      }
      g_sel[h * G_ + g] = topIdx[jb];
      blocked[jb] = true;
      for (int p = 0; p < P_; ++p) m[p] = fmaxf(m[p], Ssub[jb][p]);
    }
  }
}

// ---------------------------------------------------------------------------
// Kernel 2: main sparse attention. One 256-thread block per (bh, 16-row tile).
// Wave 0 computes the 16x48 window-score tile with V_WMMA_F32_16X16X4_F32.
// ---------------------------------------------------------------------------
__global__ void bbird_attention(const float* __restrict__ q,
                                const float* __restrict__ k,
                                const float* __restrict__ v,
                                const int* __restrict__ g_sel,
                                float* __restrict__ out) {
  __shared__ float Qs[TILE_][D_ + 1];      // +1 pad: kill LDS bank conflicts
  __shared__ float Ks[UK_][D_ + 1];
  __shared__ float rnormU[UK_];
  __shared__ float scoresU[TILE_][UK_];    // scaled q.k scores over union keys
  __shared__ float blended[TILE_][FRAG_];
  __shared__ int   ci[TILE_][KC_];
  __shared__ float rem[TILE_][KC_];
  __shared__ int   selL[TILE_][KSEL_];
  __shared__ float wL[TILE_][KSEL_ + G_];
  __shared__ float KgL[G_][D_];
  __shared__ float VgL[G_][D_];
  __shared__ float sgL[TILE_][G_];
  __shared__ int   jsel[TILE_];

  const int bh = blockIdx.y;
  const int t0 = blockIdx.x * TILE_;
  const int h  = bh / B_;
  const int tid = threadIdx.x;

  int base = t0 - FRAG_ / 2;
  if (base < 0) base = 0;
  if (base > T_ - FRAG_) base = T_ - FRAG_;

  // --- cooperative loads into LDS ---
  for (int i = tid; i < TILE_ * D_; i += blockDim.x) {
    const int r = i >> 6, d = i & 63;
    Qs[r][d] = q[((size_t)bh * T_ + t0 + r) * D_ + d];
  }
  for (int i = tid; i < UK_ * D_; i += blockDim.x) {
    const int r = i >> 6, d = i & 63;
    const int kk = base + r;
    Ks[r][d] = (kk < T_) ? k[((size_t)bh * T_ + kk) * D_ + d] : 0.f;
  }
  if (tid < G_ * D_) {
    const int g = tid >> 6, d = tid & 63;
    const int ga = g_sel[h * G_ + g];
    KgL[g][d] = k[((size_t)bh * T_ + ga) * D_ + d];
    VgL[g][d] = v[((size_t)bh * T_ + ga) * D_ + d];
  }
  __syncthreads();

  // --- reciprocal norms of union keys (shared across the 16 rows) ---
  if (tid < UK_) {
    float ss = 0.f;
    for (int d = 0; d < D_; ++d) { const float x = Ks[tid][d]; ss += x * x; }
    rnormU[tid] = 1.0f / fmaxf(sqrtf(ss), 1e-6f);
  }

  // --- wave 0: 16x48 score tile via fp32 WMMA (3 col-tiles x 16 K-steps) ---
  if (tid < 32) {
    const int lane = tid;
    const int hi = lane >> 4;         // lane[4]
    const int lr = lane & 15;         // M (A) / N (B,C)
    for (int c = 0; c < 3; ++c) {
      v8f acc = {0.f, 0.f, 0.f, 0.f, 0.f, 0.f, 0.f, 0.f};
#pragma unroll
      for (int s = 0; s < 16; ++s) {
        const int e = 4 * s + 2 * hi; // K-pair selected by lane[4] (ISA 7.12.2)
        v2f a = { Qs[lr][e],           Qs[lr][e + 1] };
        v2f b = { Ks[c * 16 + lr][e],  Ks[c * 16 + lr][e + 1] };
        acc = __builtin_amdgcn_wmma_f32_16x16x4_f32(
            false, a, false, b, (short)0, acc, false, false);
      }
#pragma unroll
      for (int r = 0; r < 8; ++r)     // C layout: VGPR r -> M = r + 8*lane[4]
        scoresU[r + 8 * hi][c * 16 + lr] = acc[r] * SCALE_;
    }
  }
  __syncthreads();

  // --- per-row selection: 16 lanes cooperate on each row ---
  const int row = tid >> 4;
  const int l16 = tid & 15;
  const int t = t0 + row;
  int start = t - FRAG_ / 2;
  if (start < 0) start = 0;
  if (start > T_ - FRAG_) start = T_ - FRAG_;
  const int off = start - base;       // in [0, 15]

  // blended = scaled score + ALPHA * distance prior
  for (int f = l16; f < FRAG_; f += 16) {
    const float dist = fabsf((float)(start + f - t));
    blended[row][f] = scoresU[row][off + f] + ALPHA_ * expf(-dist / TAU_);
  }
  __syncthreads();

  // top-24 of blended (stable selection sort == lax.top_k tie-breaking)
  if (l16 == 0) {
    float tmp[FRAG_];
    for (int f = 0; f < FRAG_; ++f) tmp[f] = blended[row][f];
    for (int r = 0; r < KC_; ++r) {
      int jb = 0; float vb = tmp[0];
      for (int f = 1; f < FRAG_; ++f) if (tmp[f] > vb) { vb = tmp[f]; jb = f; }
      ci[row][r] = jb; rem[row][r] = vb; tmp[jb] = -1e30f;
    }
  }
  __syncthreads();

  // greedy diversity steps with cosine penalties
  for (int s = 0; s < STEPS_; ++s) {
    if (l16 == 0) {
      int jb = 0; float vb = rem[row][0];
      for (int i = 1; i < KC_; ++i) if (rem[row][i] > vb) { vb = rem[row][i]; jb = i; }
      jsel[row] = jb;
      selL[row][s] = ci[row][jb];
      rem[row][jb] = NEG_;
    }
    __syncthreads();
    if (s < STEPS_ - 1) {
      const int uj = off + ci[row][jsel[row]];
      const float rnj = rnormU[uj];
      for (int i = l16; i < KC_; i += 16) {
        const int ui = off + ci[row][i];
        float dot = 0.f;
        for (int d = 0; d < D_; ++d) dot += Ks[ui][d] * Ks[uj][d];
        const float cos = dot * rnormU[ui] * rnj;
        rem[row][i] -= GAMMA_ * fmaxf(cos, 0.f);
      }
      __syncthreads();
    }
  }

  // remaining KSEL-STEPS picks = top-3 of rem
  if (l16 == 0) {
    for (int r = 0; r < KSEL_ - STEPS_; ++r) {
      int jb = 0; float vb = rem[row][0];
      for (int i = 1; i < KC_; ++i) if (rem[row][i] > vb) { vb = rem[row][i]; jb = i; }
      selL[row][STEPS_ + r] = ci[row][jb];
      rem[row][jb] = -1e30f;
    }
  }

  // global-key scores for this row
  if (l16 < G_) {
    float dot = 0.f;
    for (int d = 0; d < D_; ++d) dot += Qs[row][d] * KgL[l16][d];
    sgL[row][l16] = dot * SCALE_;
  }
  __syncthreads();

  // softmax over 6 selected + 3 global
  if (l16 == 0) {
    float sv[KSEL_ + G_];
    for (int kk = 0; kk < KSEL_; ++kk) sv[kk] = scoresU[row][off + selL[row][kk]];
    for (int g = 0; g < G_; ++g) sv[KSEL_ + g] = sgL[row][g];
    float mx = sv[0];
    for (int i = 1; i < KSEL_ + G_; ++i) mx = fmaxf(mx, sv[i]);
    float sum = 0.f;
    for (int i = 0; i < KSEL_ + G_; ++i) { sv[i] = expf(sv[i] - mx); sum += sv[i]; }
    const float inv = 1.0f / sum;
    for (int i = 0; i < KSEL_ + G_; ++i) wL[row][i] = sv[i] * inv;
  }
  __syncthreads();

  // weighted combine: 4 output dims per lane, float4 gathers of V
  {
    const int d0 = l16 * 4;
    float4 acc = make_float4(0.f, 0.f, 0.f, 0.f);
    for (int kk = 0; kk < KSEL_; ++kk) {
      const int abs_i = start + selL[row][kk];
      const float4 vv =
          *(const float4*)(v + ((size_t)bh * T_ + abs_i) * D_ + d0);
      const float wk = wL[row][kk];
      acc.x += wk * vv.x; acc.y += wk * vv.y;
      acc.z += wk * vv.z; acc.w += wk * vv.w;
    }
    for (int g = 0; g < G_; ++g) {
      const float wg = wL[row][KSEL_ + g];
      acc.x += wg * VgL[g][d0 + 0]; acc.y += wg * VgL[g][d0 + 1];
      acc.z += wg * VgL[g][d0 + 2]; acc.w += wg * VgL[g][d0 + 3];
    }
    *(float4*)(out + ((size_t)bh * T_ + t) * D_ + d0) = acc;
  }
}

// ---------------------------------------------------------------------------
extern "C" void kernel_launch(void* const* d_in, const int* in_sizes, int n_in,
                              void* d_out, int out_size, void* d_ws, size_t ws_size,
                              hipStream_t stream) {
  const float* q = (const float*)d_in[0];
  const float* k = (const float*)d_in[1];
  const float* v = (const float*)d_in[2];
  float* out = (float*)d_out;
  int* g_sel = (int*)d_ws;   // H_*G_ = 36 ints

  bbird_choose_globals<<<H_, 256, 0, stream>>>(q, k, g_sel);
  bbird_attention<<<dim3(T_ / TILE_, BH_), 256, 0, stream>>>(q, k, v, g_sel, out);
}